// ProposalTargetLayer_31636729103204
// MI455X (gfx1250) — compile-verified
//
#include <hip/hip_runtime.h>
#include <stdint.h>

#define NPROP 16384
#define NGT   128
#define NB    16
#define NT    1024
#define NSAMP 256
#define NFG   64
#define EPT   16          // elements per thread in the per-batch kernel

typedef int v4i __attribute__((ext_vector_type(4)));
typedef __attribute__((address_space(1))) v4i* gptr4_t;
typedef __attribute__((address_space(3))) v4i* lptr4_t;

// ---- async global->LDS 16B copy (CDNA5 ASYNCcnt path) ---------------------
__device__ __forceinline__ void async_b128(const void* gsrc, void* ldst) {
#if __has_builtin(__builtin_amdgcn_global_load_async_to_lds_b128)
  gptr4_t gp = (gptr4_t)(unsigned long long)gsrc;
  lptr4_t lp = (lptr4_t)(unsigned)(unsigned long long)ldst;
  __builtin_amdgcn_global_load_async_to_lds_b128(gp, lp, 0, 0);
#else
  unsigned ldsOff = (unsigned)(unsigned long long)ldst;
  unsigned long long ga = (unsigned long long)gsrc;
  asm volatile("global_load_async_to_lds_b128 %0, %1, off"
               :: "v"(ldsOff), "v"(ga) : "memory");
#endif
}
__device__ __forceinline__ void wait_async() {
  asm volatile("s_wait_asynccnt 0" ::: "memory");
}

// ---------------- JAX threefry2x32 (key = (0,1) from jax.random.key(1)) ----
__device__ __forceinline__ unsigned rotl32(unsigned x, unsigned r) {
  return (x << r) | (x >> (32u - r));
}
__device__ __forceinline__ void threefry01(unsigned x0, unsigned x1,
                                           unsigned& o0, unsigned& o1) {
  const unsigned ks0 = 0u, ks1 = 1u, ks2 = 0x1BD11BDBu;
  x0 += ks0; x1 += ks1;
#define TFR(r) { x0 += x1; x1 = rotl32(x1, (r)); x1 ^= x0; }
  TFR(13) TFR(15) TFR(26) TFR(6)   x0 += ks1; x1 += ks2 + 1u;
  TFR(17) TFR(29) TFR(16) TFR(24)  x0 += ks2; x1 += ks0 + 2u;
  TFR(13) TFR(15) TFR(26) TFR(6)   x0 += ks0; x1 += ks1 + 3u;
  TFR(17) TFR(29) TFR(16) TFR(24)  x0 += ks1; x1 += ks2 + 4u;
  TFR(13) TFR(15) TFR(26) TFR(6)   x0 += ks2; x1 += ks0 + 5u;
#undef TFR
  o0 = x0; o1 = x1;
}
__device__ __forceinline__ float jax_uniform(unsigned b, unsigned n) {
  const unsigned half = (NB * NPROP) / 2u;   // 131072
  unsigned flat = b * NPROP + n;
  unsigned c0, c1, o0, o1;
  if (flat < half) { c0 = flat; c1 = flat + half; }
  else             { c0 = flat - half; c1 = flat; }
  threefry01(c0, c1, o0, o1);
  unsigned bits = (flat < half) ? o0 : o1;
  return __uint_as_float((bits >> 9) | 0x3F800000u) - 1.0f;
}

// ---- IoU -> meta for one proposal: meta = (label+1) | (argmax_gt << 2) ----
__device__ __forceinline__ unsigned compute_meta_one(const float* __restrict__ propB,
                                                     unsigned n,
                                                     const float* gts,
                                                     const float* gareas) {
  float px1 = propB[n * 4 + 0], py1 = propB[n * 4 + 1];
  float px2 = propB[n * 4 + 2], py2 = propB[n * 4 + 3];
  float ap = (px2 - px1) * (py2 - py1);
  float best = -1.0f; unsigned bj = 0;
  for (unsigned j = 0; j < NGT; ++j) {
    float gx1 = gts[j * 4 + 0], gy1 = gts[j * 4 + 1];
    float gx2 = gts[j * 4 + 2], gy2 = gts[j * 4 + 3];
    float ix1 = fmaxf(px1, gx1), iy1 = fmaxf(py1, gy1);
    float ix2 = fminf(px2, gx2), iy2 = fminf(py2, gy2);
    float inter = fmaxf(ix2 - ix1, 0.0f) * fmaxf(iy2 - iy1, 0.0f);
    float iou = inter / fmaxf(ap + gareas[j] - inter, 1e-8f);  // IEEE div: exact thresholds/argmax
    if (iou > best) { best = iou; bj = j; }                    // first-max == jnp.argmax
  }
  unsigned code = (best >= 0.5f) ? 2u : ((best < 0.1f) ? 1u : 0u);  // label+1
  return code | (bj << 2);
}

// ---- LDS radix-select: exact k-th largest of 16384 UNIQUE u64 keys --------
__device__ unsigned long long radix_select_kth(const unsigned long long* skey,
                                               unsigned k, unsigned tid,
                                               unsigned* hist, unsigned* bc) {
  unsigned long long prefix = 0ull, pmask = 0ull;
  unsigned kk = k;
  for (int d = 7; d >= 0; --d) {
    if (tid < 256) hist[tid] = 0u;
    __syncthreads();
    for (unsigned e = 0; e < EPT; ++e) {
      unsigned long long key = skey[tid * EPT + e];
      if ((key & pmask) == prefix)
        atomicAdd(&hist[(unsigned)((key >> (8 * d)) & 255ull)], 1u);
    }
    __syncthreads();
    if (tid == 0) {
      unsigned cum = 0, dig = 0, kkN = 0;
      for (int c = 255; c >= 0; --c) {
        cum += hist[c];
        if (cum >= kk) { dig = (unsigned)c; kkN = kk - (cum - hist[c]); break; }
      }
      bc[0] = dig; bc[1] = kkN;
    }
    __syncthreads();
    unsigned dig = bc[0]; kk = bc[1];
    prefix |= ((unsigned long long)dig) << (8 * d);
    pmask  |= ((unsigned long long)0xFFull) << (8 * d);
    __syncthreads();   // keep hist re-zero of next pass ordered after bc reads
  }
  return prefix;       // the k-th largest key itself (keys unique)
}

// ---- small bitonic: sort 256 u64 ascending --------------------------------
__device__ __forceinline__ void sort256(unsigned long long* a, unsigned tid) {
  for (unsigned k = 2; k <= NSAMP; k <<= 1) {
    for (unsigned j = k >> 1; j > 0; j >>= 1) {
      __syncthreads();
      if (tid < NSAMP / 2) {
        unsigned i = (tid << 1) - (tid & (j - 1));
        unsigned p = i + j;
        bool up = ((i & k) == 0);
        unsigned long long x = a[i], y = a[p];
        if (up ? (x > y) : (x < y)) { a[i] = y; a[p] = x; }
      }
    }
  }
  __syncthreads();
}

// ============ K1: IoU -> meta, one proposal per thread (256 blocks) ========
__global__ __launch_bounds__(NT)
void iou_meta_kernel(const float* __restrict__ proposals,
                     const float* __restrict__ gt,
                     unsigned* __restrict__ meta_out) {
  __shared__ float gts[NGT * 4];
  __shared__ float gareas[NGT];
  const unsigned tid = threadIdx.x;
  const unsigned b = blockIdx.x >> 4;
  const unsigned chunk = blockIdx.x & 15;

  if (tid < NGT) async_b128(gt + ((size_t)b * NGT + tid) * 4, &gts[tid * 4]);
  wait_async();
  __syncthreads();
  if (tid < NGT)
    gareas[tid] = (gts[tid * 4 + 2] - gts[tid * 4 + 0]) *
                  (gts[tid * 4 + 3] - gts[tid * 4 + 1]);
  __syncthreads();

  const float* propB = proposals + (size_t)b * NPROP * 4;
  unsigned n = chunk * NT + tid;
  meta_out[(size_t)b * NPROP + n] = compute_meta_one(propB, n, gts, gareas);
}

// ============ K2: per-batch selection + outputs (16 blocks) ================
__global__ __launch_bounds__(NT)
void select_kernel(const float* __restrict__ proposals,
                   const float* __restrict__ gt,
                   const unsigned* __restrict__ meta_in,
                   int use_meta,
                   float* __restrict__ out) {
  __shared__ unsigned long long skey[NPROP];      // 128 KB
  __shared__ unsigned smeta[NPROP];               //  64 KB
  __shared__ float    gts[NGT * 4];               //   2 KB
  __shared__ float    gareas[NGT];                //  .5 KB
  __shared__ unsigned posSel[NPROP / 32];         //   2 KB
  __shared__ unsigned hist[256];                  //   1 KB
  __shared__ unsigned long long selBuf[NSAMP];    //   2 KB
  __shared__ unsigned bc[2];
  __shared__ unsigned sNpos, sSelCnt;

  const unsigned tid = threadIdx.x;
  const unsigned b = blockIdx.x;
  const float* propB = proposals + (size_t)b * NPROP * 4;
  const unsigned n0 = tid * EPT;

  if (tid == 0) { sNpos = 0u; sSelCnt = 0u; }
  if (tid < NPROP / 32) posSel[tid] = 0u;

  // ---- async staging: gt tile (+ meta tile if precomputed by K1) ----
  if (tid < NGT) async_b128(gt + ((size_t)b * NGT + tid) * 4, &gts[tid * 4]);
  if (use_meta) {
    const unsigned* msrc = meta_in + (size_t)b * NPROP;
    for (unsigned i = 0; i < 4; ++i) {          // 4096 x 16B = 64 KB
      unsigned l = i * NT + tid;
      async_b128(msrc + l * 4, &smeta[l * 4]);
    }
  }
  wait_async();
  __syncthreads();
  if (tid < NGT)
    gareas[tid] = (gts[tid * 4 + 2] - gts[tid * 4 + 0]) *
                  (gts[tid * 4 + 3] - gts[tid * 4 + 1]);
  __syncthreads();

  if (!use_meta) {                               // fallback: compute meta here
    for (unsigned e = 0; e < EPT; ++e) {
      unsigned n = n0 + e;
      smeta[n] = compute_meta_one(propB, n, gts, gareas);
    }
  }
  __syncthreads();

  // ---- key1 = rank positives by r (ties -> lower index); cache r ----
  float rloc[EPT];
  unsigned myPos = 0;
  for (unsigned e = 0; e < EPT; ++e) {
    unsigned n = n0 + e;
    unsigned code = smeta[n] & 3u;
    float r = jax_uniform(b, n);
    rloc[e] = r;
    if (code == 2u) myPos++;
    unsigned hi = ((code == 2u) ? (1u << 30) : 0u) | __float_as_uint(r);
    skey[n] = (((unsigned long long)hi) << 32) | (unsigned)(~n);
  }
  if (myPos) atomicAdd(&sNpos, myPos);
  __syncthreads();

  // ---- pos_sel = top min(64, npos) positives by r (membership only) ----
  unsigned npos = sNpos;
  unsigned k1 = (npos < NFG) ? npos : NFG;
  if (k1 > 0) {
    unsigned long long thr1 = radix_select_kth(skey, k1, tid, hist, bc);
    for (unsigned e = 0; e < EPT; ++e) {
      unsigned n = n0 + e;
      if (skey[n] >= thr1) atomicOr(&posSel[n >> 5], 1u << (n & 31));
    }
  }
  __syncthreads();

  // ---- key2 = jax.lax.top_k priority ordering ----
  for (unsigned e = 0; e < EPT; ++e) {
    unsigned n = n0 + e;
    unsigned code = smeta[n] & 3u;
    float r = rloc[e];
    bool psel = (posSel[n >> 5] >> (n & 31)) & 1u;
    float prio = psel ? (3.0f + r)
               : (code == 1u ? (2.0f + r)
               : (code == 2u ? (1.0f + r) : r));
    skey[n] = (((unsigned long long)__float_as_uint(prio)) << 32) | (unsigned)(~n);
  }
  __syncthreads();

  // ---- exact top-256: radix threshold, compact, sort the 256 winners ----
  unsigned long long thr2 = radix_select_kth(skey, NSAMP, tid, hist, bc);
  for (unsigned e = 0; e < EPT; ++e) {
    unsigned n = n0 + e;
    unsigned long long key = skey[n];
    if (key >= thr2) {
      unsigned p = atomicAdd(&sSelCnt, 1u);
      selBuf[p] = ~key;                          // ascending sort of ~key == desc of key
    }
  }
  sort256(selBuf, tid);

  // ---- emit ----
  if (tid < NSAMP) {
    unsigned n = (unsigned)(selBuf[tid] & 0xFFFFFFFFull);   // low32(~key2) == n
    unsigned m = smeta[n];
    int bl = ((m & 3u) == 2u) ? 1 : 0;                      // max(label, 0)
    float px1 = propB[n * 4 + 0], py1 = propB[n * 4 + 1];
    float px2 = propB[n * 4 + 2], py2 = propB[n * 4 + 3];
    unsigned gj = (m >> 2) & 0x7Fu;
    float gx1 = gts[gj * 4 + 0], gy1 = gts[gj * 4 + 1];
    float gx2 = gts[gj * 4 + 2], gy2 = gts[gj * 4 + 3];

    float rx = (px1 + px2) * 0.5f, ry = (py1 + py2) * 0.5f;
    float rw = px2 - px1,          rh = py2 - py1;
    float gx = (gx1 + gx2) * 0.5f, gy = (gy1 + gy2) * 0.5f;
    float gw = gx2 - gx1,          gh = gy2 - gy1;
    float t0 = (gx - rx) / rw;
    float t1 = (gy - ry) / rh;
    float t2 = logf(gw / rw);
    float t3 = logf(gh / rh);
    float w  = bl ? 1.0f : 0.0f;

    const size_t OUT_ROI = 0;       // (16,256,4)
    const size_t OUT_LAB = 16384;   // (16,256)
    const size_t OUT_TGT = 20480;   // (16,256,4)
    const size_t OUT_INW = 36864;   // (16,256,4)
    const size_t OUT_OUW = 53248;   // (16,256,4)
    size_t o = (size_t)b * NSAMP + tid;

    out[OUT_ROI + o * 4 + 0] = px1;
    out[OUT_ROI + o * 4 + 1] = py1;
    out[OUT_ROI + o * 4 + 2] = px2;
    out[OUT_ROI + o * 4 + 3] = py2;
    out[OUT_LAB + o] = (float)bl;
    out[OUT_TGT + o * 4 + 0] = t0;
    out[OUT_TGT + o * 4 + 1] = t1;
    out[OUT_TGT + o * 4 + 2] = t2;
    out[OUT_TGT + o * 4 + 3] = t3;
    out[OUT_INW + o * 4 + 0] = w; out[OUT_INW + o * 4 + 1] = w;
    out[OUT_INW + o * 4 + 2] = w; out[OUT_INW + o * 4 + 3] = w;
    out[OUT_OUW + o * 4 + 0] = w; out[OUT_OUW + o * 4 + 1] = w;
    out[OUT_OUW + o * 4 + 2] = w; out[OUT_OUW + o * 4 + 3] = w;
  }
}

extern "C" void kernel_launch(void* const* d_in, const int* in_sizes, int n_in,
                              void* d_out, int out_size, void* d_ws, size_t ws_size,
                              hipStream_t stream) {
  (void)in_sizes; (void)n_in; (void)out_size;
  const float* proposals = (const float*)d_in[0];
  const float* gt_boxes  = (const float*)d_in[1];
  float* out = (float*)d_out;

  const size_t metaBytes = (size_t)NB * NPROP * sizeof(unsigned);
  const int use_meta = (d_ws != nullptr && ws_size >= metaBytes) ? 1 : 0;

  if (use_meta) {
    // 256 small blocks: fills the whole GPU for the IoU phase
    iou_meta_kernel<<<dim3(NB * 16), dim3(NT), 0, stream>>>(
        proposals, gt_boxes, (unsigned*)d_ws);
  }
  select_kernel<<<dim3(NB), dim3(NT), 0, stream>>>(
      proposals, gt_boxes, (const unsigned*)d_ws, use_meta, out);
}